// BITypeNetwork_27908697489416
// MI455X (gfx1250) — compile-verified
//
#include <hip/hip_runtime.h>

// CDNA5 / gfx1250. wave32. Core op: 16384x16384 fp32 GEMV mapped onto
// V_WMMA_F32_16X16X4_F32, memory-bound at ~1.07 GB of adj per call
// (roofline ~46 us @ 23.3 TB/s).

#define NNEUR 16384

typedef __attribute__((ext_vector_type(2))) float v2f;
typedef __attribute__((ext_vector_type(8))) float v8f;

__global__ __launch_bounds__(256) void binet_wmma_kernel(
    const float* __restrict__ x,
    const float* __restrict__ adj,
    const float* __restrict__ states,
    const float* __restrict__ c,
    float* __restrict__ out)
{
    __shared__ float partial[8 * 16];   // [wave][row-in-block]

    const int tid  = threadIdx.x;
    const int wave = tid >> 5;          // 0..7
    const int lane = tid & 31;          // 0..31
    const int r0   = blockIdx.x << 4;   // 16 rows per block

    // f32 16x4 A-fragment layout: M = lane%16; lanes 0-15 hold K=(0,1),
    // lanes 16-31 hold K=(2,3) -> one contiguous b64 load per lane.
    const int mRow = r0 + (lane & 15);
    const int cOff = (lane < 16) ? 0 : 2;

    const float* __restrict__ arow = adj + (size_t)mRow * NNEUR + cOff;
    const float* __restrict__ srow = states + cOff;

    // Each wave covers a 2048-wide K slice: 512 wmma ops, 2 accumulators
    // to break the D->C dependency chain.
    v8f acc0 = {};
    v8f acc1 = {};
    const int kBeg = wave * (NNEUR / 8);
    const int kEnd = kBeg + (NNEUR / 8);

#pragma unroll 2
    for (int k = kBeg; k < kEnd; k += 8) {
        v2f a0 = *(const v2f*)(arow + k);
        v2f s0 = *(const v2f*)(srow + k);
        v2f b0;
        b0.x = 1.0f - s0.x;             // w[j] = 1 - states[j], exact 0/1
        b0.y = 1.0f - s0.y;
        acc0 = __builtin_amdgcn_wmma_f32_16x16x4_f32(
            false, a0, false, b0, (short)0, acc0, false, false);

        v2f a1 = *(const v2f*)(arow + k + 4);
        v2f s1 = *(const v2f*)(srow + k + 4);
        v2f b1;
        b1.x = 1.0f - s1.x;
        b1.y = 1.0f - s1.y;
        acc1 = __builtin_amdgcn_wmma_f32_16x16x4_f32(
            false, a1, false, b1, (short)0, acc1, false, false);
    }

    v8f acc = acc0 + acc1;

    // D layout: VGPR i holds row M=i (lanes 0-15) and M=8+i (lanes 16-31);
    // all 16 columns are identical (B was K-broadcast), so lane 0 / lane 16
    // each publish 8 row partials.
    if (lane == 0) {
#pragma unroll
        for (int i = 0; i < 8; ++i) partial[wave * 16 + i] = acc[i];
    } else if (lane == 16) {
#pragma unroll
        for (int i = 0; i < 8; ++i) partial[wave * 16 + 8 + i] = acc[i];
    }
    __syncthreads();

    if (tid < 16) {
        float y = 0.0f;                 // exact integer count: #zero-state neighbors
#pragma unroll
        for (int wv = 0; wv < 8; ++wv) y += partial[wv * 16 + tid];

        const int i = r0 + tid;
        const float inter = (y != 0.0f) ? 1.0f : 0.0f;          // 1 - prod
        const float x3    = x[(i + 1) & (NNEUR - 1)];
        const float cnew  = 1.0f - c[i] * x3;
        out[i] = 1.0f - cnew * inter;
    }
}

extern "C" void kernel_launch(void* const* d_in, const int* in_sizes, int n_in,
                              void* d_out, int out_size, void* d_ws, size_t ws_size,
                              hipStream_t stream) {
    (void)in_sizes; (void)n_in; (void)d_ws; (void)ws_size; (void)out_size;
    const float* x      = (const float*)d_in[0];
    const float* adj    = (const float*)d_in[1];
    const float* states = (const float*)d_in[2];
    const float* c      = (const float*)d_in[3];
    float* out          = (float*)d_out;

    binet_wmma_kernel<<<NNEUR / 16, 256, 0, stream>>>(x, adj, states, c, out);
}